// SAID_monomer_74826920231208
// MI455X (gfx1250) — compile-verified
//
#include <hip/hip_runtime.h>
#include <hip/hip_bf16.h>

// ---------------------------------------------------------------------------
// SAID / Fastfood projection for MI455X (gfx1250, wave32, WMMA + TDM).
//   out = init + said_scale * FWHT(perm_gather(FWHT(pad(iv)*B)) * G) / sqrt(LL*sum(G^2))
// LL = 2^25 == DD:
//   * first FWHT == FWHT_4096 tiled  ->  w[4096] (one-workgroup LDS kernel)
//   * gather:      t[i] = w[Pi[i] & 4095] * G[i]          (B128 streams)
//   * second FWHT: three radix-256 WMMA passes (bits 0-7, 8-15, 16-23),
//                  tiles DMA'd into LDS by the Tensor Data Mover with
//                  hardware pad-on-load giving the 33-float LDS pitch;
//                  write-back as global_store_b128
//   * top-bit radix-2 fused into the finalize kernel (B128 streams)
// t lives in d_out (passes are in-place); d_ws holds only ~21KB of scalars.
// ---------------------------------------------------------------------------

#define LLN      (1u << 25)
#define HALF_LLN (1u << 24)

typedef __attribute__((ext_vector_type(2))) float    v2f;
typedef __attribute__((ext_vector_type(4))) float    v4f;
typedef __attribute__((ext_vector_type(8))) float    v8f;
typedef __attribute__((ext_vector_type(4))) unsigned v4u;
typedef __attribute__((ext_vector_type(4))) int      v4i;
typedef __attribute__((ext_vector_type(8))) int      v8i;

#if __has_include(<hip/amd_detail/amd_gfx1250_TDM.h>)
#define TDM_ARGS6 1          // therock/clang-23 toolchain: 6-arg builtin
#else
#define TDM_ARGS6 0          // ROCm 7.2 / clang-22: 5-arg builtin
#endif

__device__ __forceinline__ float hsignf(int x) {
    // (+1.0f / -1.0f) from parity of popcount, branchless
    return __int_as_float(0x3F800000u | ((unsigned)(__popc(x) & 1) << 31));
}

// ---------------------------------------------------------------- TDM ------
// Issue a TENSOR_LOAD_TO_LDS of a tile into LDS at lds_off.
//   2D: tile_d2 == 0 : x = tile_d0 contiguous elems, y = tile_d1 rows @ stride0
//   3D: tile_d2 != 0 : + z = tile_d2 planes @ stride1
// data_size = 4 bytes; pad-on-load: +1 DWORD after every 32 DWORDs stored,
// reproducing the 33-float LDS row pitch used by the compute phase.
__device__ __forceinline__ void tdm_load_tile(const float* gptr, unsigned lds_off,
                                              unsigned tile_d0, unsigned tile_d1,
                                              unsigned tile_d2,
                                              unsigned long long stride0,
                                              unsigned long long stride1) {
    unsigned long long ga = (unsigned long long)(uintptr_t)gptr;
    const unsigned big = 0x40000000u;            // generous tensor dims (no OOB)
    v4u g0;
    g0[0] = 1u;                                   // count=1, user descriptor
    g0[1] = lds_off;                              // lds_addr (bytes)
    g0[2] = (unsigned)(ga & 0xFFFFFFFFu);         // global_addr[31:0]
    g0[3] = (unsigned)((ga >> 32) & 0x01FFFFFFu)  // global_addr[56:32]
          | 0x80000000u;                          // type = 2 ("image")
    unsigned w0 = (2u << 16)                      // data_size = 4B
                | (1u << 20)                      // pad_enable
                | (4u << 22);                     // pad_interval: 32 DWORDs (amount=1)
    unsigned w1 = (big & 0xFFFFu) << 16;                          // tensor_dim0 lo
    unsigned w2 = (big >> 16) | ((big & 0xFFFFu) << 16);          // dim0 hi | dim1 lo
    unsigned w3 = (big >> 16) | (tile_d0 << 16);                  // dim1 hi | tile_dim0
    unsigned w4 = (tile_d1 & 0xFFFFu) | (tile_d2 << 16);          // tile_dim1 | tile_dim2
    unsigned w5 = (unsigned)(stride0 & 0xFFFFFFFFu);              // dim0_stride[31:0]
    unsigned w6 = (unsigned)((stride0 >> 32) & 0xFFFFu)
                | ((unsigned)(stride1 & 0xFFFFu) << 16);          // |dim1_stride lo
    unsigned w7 = (unsigned)((stride1 >> 16) & 0xFFFFFFFFu);      // dim1_stride hi
    v8i g1 = { (int)w0, (int)w1, (int)w2, (int)w3,
               (int)w4, (int)w5, (int)w6, (int)w7 };
    v4i g2 = { (int)big, (int)big, 0, 0 };        // tensor_dim2/3, stride2=0, tile_dim3=0
    v4i g3 = { 0, 0, 0, 0 };
#if TDM_ARGS6
    v8i g4 = { 0, 0, 0, 0, 0, 0, 0, 0 };
    __builtin_amdgcn_tensor_load_to_lds(g0, g1, g2, g3, g4, 0);
#else
    __builtin_amdgcn_tensor_load_to_lds(g0, g1, g2, g3, 0);
#endif
}

// -------------------------------------------------------------- FWHT_4096 --
__global__ void __launch_bounds__(256) k_fwht4096(const float* __restrict__ iv,
                                                  const float* __restrict__ BB,
                                                  float* __restrict__ w) {
    __shared__ float x[4096];
    const int tid = threadIdx.x;
    for (int t = tid; t < 4096; t += 256) x[t] = iv[t] * BB[t];
    __syncthreads();
    for (int s = 2048; s >= 1; s >>= 1) {
        for (int p = tid; p < 2048; p += 256) {
            int i = ((p / s) * (s << 1)) + (p % s);
            float a = x[i], b = x[i + s];
            x[i]     = a + b;
            x[i + s] = a - b;
        }
        __syncthreads();
    }
    for (int t = tid; t < 4096; t += 256) w[t] = x[t];
}

// ------------------------------------------------------------ sum(G^2) ----
__global__ void __launch_bounds__(256) k_gg_partial(const float* __restrict__ GG,
                                                    float* __restrict__ partials) {
    __shared__ float red[256];
    const v4f* __restrict__ GG4 = (const v4f*)GG;
    const unsigned tid = threadIdx.x;
    const unsigned stride = gridDim.x * blockDim.x;
    float s = 0.0f;
    for (unsigned i = blockIdx.x * blockDim.x + tid; i < (LLN / 4u); i += stride) {
        v4f g = GG4[i];
        s = fmaf(g[0], g[0], s);
        s = fmaf(g[1], g[1], s);
        s = fmaf(g[2], g[2], s);
        s = fmaf(g[3], g[3], s);
    }
    red[tid] = s;
    __syncthreads();
    for (int o = 128; o > 0; o >>= 1) {
        if (tid < (unsigned)o) red[tid] += red[tid + o];
        __syncthreads();
    }
    if (tid == 0) partials[blockIdx.x] = red[0];
}

__global__ void __launch_bounds__(256) k_gg_final(const float* __restrict__ partials,
                                                  const float* __restrict__ said_scale,
                                                  float* __restrict__ alpha_out) {
    __shared__ float red[256];
    const unsigned tid = threadIdx.x;
    red[tid] = partials[tid] + partials[tid + 256] +
               partials[tid + 512] + partials[tid + 768];
    __syncthreads();
    for (int o = 128; o > 0; o >>= 1) {
        if (tid < (unsigned)o) red[tid] += red[tid + o];
        __syncthreads();
    }
    if (tid == 0) {
        alpha_out[0] = said_scale[0] / sqrtf((float)LLN * red[0]);
    }
}

// ----------------------------------------- gather: t = w[Pi & 4095] * G ----
__global__ void __launch_bounds__(256) k_build_t(const float* __restrict__ w,
                                                 const int* __restrict__ Pi,
                                                 const float* __restrict__ GG,
                                                 float* __restrict__ t) {
    __shared__ float wl[4096];
    for (int i = threadIdx.x; i < 4096; i += 256) wl[i] = w[i];
    __syncthreads();
    const v4i* __restrict__ Pi4 = (const v4i*)Pi;
    const v4f* __restrict__ GG4 = (const v4f*)GG;
    v4f*       __restrict__ t4  = (v4f*)t;
    const unsigned stride = gridDim.x * blockDim.x;
    for (unsigned i = blockIdx.x * blockDim.x + threadIdx.x; i < (LLN / 4u); i += stride) {
        if (i + stride < (LLN / 4u)) {
            __builtin_prefetch(Pi4 + i + stride, 0, 0);   // global_prefetch_b8
            __builtin_prefetch(GG4 + i + stride, 0, 0);
        }
        v4i p = Pi4[i];
        v4f g = GG4[i];
        v4f r;
        r[0] = wl[p[0] & 4095] * g[0];
        r[1] = wl[p[1] & 4095] * g[1];
        r[2] = wl[p[2] & 4095] * g[2];
        r[3] = wl[p[3] & 4095] * g[3];
        t4[i] = r;
    }
}

// --------------------------------------------- radix-256 WMMA FWHT pass ----
// Transforms bits [log2 S, log2 S + 8) of the index for every element of t.
// Vector v (v in [0,2^17)): element g lives at  base(v) + g*S,
//   base(v) = (v / S) * (256*S) + (v % S).
// Block = 32 vectors (tile 256 x 32 floats, ~33KB LDS), 8 wave32s, 4 vec/wave.
// Tile load: one TENSOR_LOAD_TO_LDS (wave 0) + s_wait_tensorcnt + barrier.
// Per vector: Y = H16 * V * H16 via 2 chained-K16 f32 WMMAs (8x 16x16x4).
// Write-back: 8x global_store_b128 per thread.
#define LPITCH 33

__global__ void __launch_bounds__(256) k_fwht256_pass(float* __restrict__ t,
                                                      unsigned S) {
    __shared__ float lds[256][LPITCH];
    const unsigned tid  = threadIdx.x;
    const unsigned lane = tid & 31;
    const unsigned wid  = tid >> 5;
    const unsigned v0   = blockIdx.x * 32;
    const unsigned base = (S == 1u) ? v0 * 256u
                                    : (v0 / S) * (S << 8) + (v0 % S);

    // ---- tile load via Tensor Data Mover (single wave issues the DMA) ----
    if (tid < 32) {
        const unsigned lds_off = (unsigned)(uintptr_t)(&lds[0][0]);
        if (S == 1u) {
            // 3D: x = 1 elem, y = 32 vectors @ stride 256, z = 256 elems @ stride 1
            // -> TDM emits the transposed (g-major) layout directly.
            tdm_load_tile(t + base, lds_off, 1u, 32u, 256u, 256ull, 1ull);
        } else {
            // 2D: x = 32 contiguous lanes, y = 256 rows @ stride S
            tdm_load_tile(t + base, lds_off, 32u, 256u, 0u,
                          (unsigned long long)S, 0ull);
        }
        __builtin_amdgcn_s_wait_tensorcnt(0);
    }
    __syncthreads();

    // ---- A = H16 column chunks, generated in registers (exact +-1) ----
    const int m   = (int)(lane & 15);
    const int khi = (int)((lane >> 4) << 1);      // K offset 0 or 2 per half-wave
    v2f A[4];
    #pragma unroll
    for (int c = 0; c < 4; ++c) {
        A[c][0] = hsignf(m & (4 * c + khi + 0));
        A[c][1] = hsignf(m & (4 * c + khi + 1));
    }
    const int n     = m;                          // frag column index
    const int mbase = (int)((lane >> 4) << 3);    // D rows 0..7 / 8..15

    for (int q = 0; q < 4; ++q) {
        const int j = (int)(wid * 4) + q;         // wave-private column
        // ---- T = H16 * V ----
        v8f acc = {};
        #pragma unroll
        for (int c = 0; c < 4; ++c) {
            const int r0 = 4 * c + khi;
            v2f B;
            B[0] = lds[r0 * 16 + n][j];
            B[1] = lds[(r0 + 1) * 16 + n][j];
            acc = __builtin_amdgcn_wmma_f32_16x16x4_f32(
                false, A[c], false, B, (short)0, acc, false, false);
        }
        // store T^T (transpose folded into the store); wave-private column,
        // same-wave DS ordering makes this safe without a barrier
        #pragma unroll
        for (int r = 0; r < 8; ++r) {
            lds[n * 16 + mbase + r][j] = acc[r];
        }
        // ---- Z = H16 * T^T ; Y = Z^T ----
        v8f acc2 = {};
        #pragma unroll
        for (int c = 0; c < 4; ++c) {
            const int r0 = 4 * c + khi;
            v2f B;
            B[0] = lds[r0 * 16 + n][j];
            B[1] = lds[(r0 + 1) * 16 + n][j];
            acc2 = __builtin_amdgcn_wmma_f32_16x16x4_f32(
                false, A[c], false, B, (short)0, acc2, false, false);
        }
        #pragma unroll
        for (int r = 0; r < 8; ++r) {
            lds[n * 16 + mbase + r][j] = acc2[r];
        }
    }
    __syncthreads();

    // ---- store tile back: 2048 float4s, 8 global_store_b128 per thread ----
    if (S == 1u) {
        #pragma unroll 2
        for (unsigned it = 0; it < 8; ++it) {
            unsigned idx = it * 256u + tid;          // [0, 2048)
            unsigned k   = idx >> 6;                 // vector j = k
            unsigned gq  = (idx & 63u) << 2;         // g base (x4)
            v4f val = { lds[gq + 0][k], lds[gq + 1][k],
                        lds[gq + 2][k], lds[gq + 3][k] };
            *(v4f*)(t + base + k * 256u + gq) = val;
        }
    } else {
        #pragma unroll 2
        for (unsigned it = 0; it < 8; ++it) {
            unsigned idx = it * 256u + tid;          // [0, 2048)
            unsigned g   = idx >> 3;                 // row
            unsigned j4  = (idx & 7u) << 2;          // vector base (x4)
            v4f val = { lds[g][j4 + 0], lds[g][j4 + 1],
                        lds[g][j4 + 2], lds[g][j4 + 3] };
            *(v4f*)(t + base + g * S + j4) = val;
        }
    }
}

// ------------------------- top-bit radix-2 + scale + add init (fused) ------
__global__ void __launch_bounds__(256) k_finalize(float* __restrict__ t,
                                                  const float* __restrict__ init,
                                                  const float* __restrict__ alpha_p) {
    const float alpha = alpha_p[0];
    v4f*       __restrict__ t4 = (v4f*)t;
    const v4f* __restrict__ i4 = (const v4f*)init;
    const unsigned stride = gridDim.x * blockDim.x;
    const unsigned H4 = HALF_LLN / 4u;
    for (unsigned i = blockIdx.x * blockDim.x + threadIdx.x; i < H4; i += stride) {
        v4f a = t4[i], b = t4[i + H4];
        v4f ia = i4[i], ib = i4[i + H4];
        v4f lo, hi;
        #pragma unroll
        for (int c = 0; c < 4; ++c) {
            lo[c] = fmaf(alpha, a[c] + b[c], ia[c]);
            hi[c] = fmaf(alpha, a[c] - b[c], ib[c]);
        }
        t4[i]      = lo;
        t4[i + H4] = hi;
    }
}

// ---------------------------------------------------------------------------
extern "C" void kernel_launch(void* const* d_in, const int* in_sizes, int n_in,
                              void* d_out, int out_size, void* d_ws, size_t ws_size,
                              hipStream_t stream) {
    const float* iv   = (const float*)d_in[0];   // intrinsic_vec [4096]
    const float* BB   = (const float*)d_in[1];   // [2^25]
    const float* GG   = (const float*)d_in[2];   // [2^25]
    const float* init = (const float*)d_in[3];   // [4096*8192]
    const float* said = (const float*)d_in[4];   // [1]
    const int*   Pi   = (const int*)d_in[5];     // [2^25]
    float* out = (float*)d_out;                  // also the in-place work vector t

    float* wsf      = (float*)d_ws;
    float* w        = wsf;                       // 4096
    float* partials = wsf + 4096;                // 1024
    float* alpha    = wsf + 4096 + 1024;         // 1

    k_fwht4096   <<<1,    256, 0, stream>>>(iv, BB, w);
    k_gg_partial <<<1024, 256, 0, stream>>>(GG, partials);
    k_gg_final   <<<1,    256, 0, stream>>>(partials, said, alpha);
    k_build_t    <<<4096, 256, 0, stream>>>(w, Pi, GG, out);
    // second FWHT: 2^17 vectors / 32 per block = 4096 blocks per pass
    k_fwht256_pass<<<4096, 256, 0, stream>>>(out, 1u);       // bits 0-7
    k_fwht256_pass<<<4096, 256, 0, stream>>>(out, 256u);     // bits 8-15
    k_fwht256_pass<<<4096, 256, 0, stream>>>(out, 65536u);   // bits 16-23
    k_finalize   <<<8192, 256, 0, stream>>>(out, init, alpha); // bit 24 + scale + init
}